// QLSTM_65481071402308
// MI455X (gfx1250) — compile-verified
//
#include <hip/hip_runtime.h>
#include <math.h>

// QLSTM (degenerate: h_prev/c_prev constant per layer) on MI455X / gfx1250.
// Per layer: gates[T*B, 4H] = X[T*B,256] @ W[256,4H] + hterm, fused gate
// activations in the epilogue. f32 WMMA (V_WMMA_F32_16X16X4_F32) keeps the
// math exact vs the f32 reference; ~137 GFLOP roughly balances ~0.8 GiB of
// HBM traffic (~35us at 23.3 TB/s), so staying in f32 is roofline-optimal.
//
// v2: B slab stored TRANSPOSED in LDS so each B fragment is one aligned
// ds_load_b64 (kills the v_dual_mov repacking seen in v1), and wave tile
// rebalanced to 2 A-frags x 4 B-frags => 6 ds_load_b64 per 8 WMMA.

typedef __attribute__((ext_vector_type(2))) float v2f;
typedef __attribute__((ext_vector_type(8))) float v8f;

#define T_SEQ   2048
#define BATCH   64
#define DIMIN   256
#define HID     256
#define NLAYER  2
#define KTOT    256                 // == DIMIN == HID
#define MTOT    (T_SEQ * BATCH)     // 131072 rows
#define TBH     ((size_t)T_SEQ * BATCH * HID)            // 33554432
#define TLBH    ((size_t)T_SEQ * NLAYER * BATCH * HID)   // 67108864

// Block tiling
#define MT 64          // rows per block
#define OT 64          // o-columns per block (block computes all 4 gates)
#define KT 32          // K slab staged through LDS
#define A_STRIDE 36    // 32 + pad: float4-aligned, conflict-free frag reads
#define BT_STRIDE 34   // KT + pad: even (b64-aligned), conflict-free reads

// ---------------------------------------------------------------------------
// hterm[l,b,g,o] = bias[l,g,o] + sum_j h0[l,b,j] * W[l,g,DIMIN+j,o]
// (h_0 is constant over time in the reference, so this folds into a bias.)
// ---------------------------------------------------------------------------
__global__ __launch_bounds__(256)
void lstm_hterm(const float* __restrict__ h0, const float* __restrict__ W,
                const float* __restrict__ bias, float* __restrict__ hterm) {
    int idx = blockIdx.x * 256 + threadIdx.x;   // [L, B, 4, H] flat, 131072
    int o = idx & (HID - 1);
    int g = (idx >> 8) & 3;
    int b = (idx >> 10) & (BATCH - 1);
    int l = idx >> 16;
    const float* h = h0 + (size_t)(l * BATCH + b) * HID;
    const float* w = W + ((size_t)(l * 4 + g) * (DIMIN + HID) + DIMIN) * HID + o;
    float acc = bias[(l * 4 + g) * HID + o];
    for (int j = 0; j < HID; ++j) acc += h[j] * w[(size_t)j * HID];
    hterm[idx] = acc;
}

// ---------------------------------------------------------------------------
// One LSTM layer: gates GEMM via f32 WMMA + fused activations.
// A row r = t*BATCH + b lives at A + t*a_stride_t + b*256 (covers both the
// contiguous input tensor and the strided h_n[:,0] region of d_out).
// ---------------------------------------------------------------------------
__global__ __launch_bounds__(256)
void lstm_layer_wmma(const float* A, int a_stride_t,
                     const float* __restrict__ Wl,       // W + l*4*(D+H)*H
                     const float* __restrict__ hterm_l,  // [B,4,H]
                     const float* __restrict__ c0_l,     // [B,H]
                     float* out, int layer) {
    __shared__ float As[MT][A_STRIDE];          //  9.0 KB, row-major [m][k]
    __shared__ float Bt[4][OT][BT_STRIDE];      // 34.0 KB, TRANSPOSED [g][n][k]

    const int tid   = threadIdx.x;
    const int lane  = tid & 31;
    const int wave  = tid >> 5;
    const int waveM = wave & 1;     // 2 M-slots of 32 rows
    const int waveN = wave >> 1;    // 4 N-slots of 16 o-cols
    const int m0 = blockIdx.x * MT; // global row base
    const int o0 = blockIdx.y * OT; // global o base

    v8f acc[4][2];                  // [gate][mi], 64 VGPRs
    #pragma unroll
    for (int g = 0; g < 4; ++g)
        #pragma unroll
        for (int mi = 0; mi < 2; ++mi) acc[g][mi] = (v8f)0.f;

    const int lm = lane & 15;
    const int lh = lane >> 4;       // selects K pair per ISA 16x4 A/B layout

    for (int k0 = 0; k0 < KTOT; k0 += KT) {
        // --- stage A slab: 64 rows x 32 K = 512 float4, 2 per thread -------
        #pragma unroll
        for (int it = 0; it < 2; ++it) {
            int f = tid + it * 256;         // 0..511
            int row = f >> 3;
            int kk = (f & 7) * 4;
            int r = m0 + row;
            int t = r >> 6;                 // r / BATCH
            int b = r & (BATCH - 1);
            const float* src = A + (size_t)t * a_stride_t + b * 256 + k0 + kk;
            *(float4*)&As[row][kk] = *(const float4*)src;
            if (k0 + KT < KTOT) __builtin_prefetch(src + KT, 0, 1);
        }
        // --- stage B slabs transposed: 4 gates x 32K x 64o, 8 float4/thread
        #pragma unroll
        for (int it = 0; it < 8; ++it) {
            int f = tid + it * 256;         // 0..2047
            int g = f >> 9;
            int f2 = f & 511;
            int krow = f2 >> 4;             // K row in slab (32 rows)
            int nn = (f2 & 15) * 4;         // o column quad
            const float* src = Wl + (size_t)g * (DIMIN + HID) * HID
                                  + (size_t)(k0 + krow) * HID + o0 + nn;
            float4 v = *(const float4*)src;
            Bt[g][nn + 0][krow] = v.x;      // transpose on the way in
            Bt[g][nn + 1][krow] = v.y;
            Bt[g][nn + 2][krow] = v.z;
            Bt[g][nn + 3][krow] = v.w;
            if (k0 + KT < KTOT) __builtin_prefetch(src + (size_t)KT * HID, 0, 1);
        }
        __syncthreads();

        // --- 8 K-steps: 6 ds_load_b64 + 8 WMMA each ------------------------
        #pragma unroll
        for (int kk = 0; kk < KT; kk += 4) {
            // frag pair K=kk+2lh, kk+2lh+1 : contiguous -> single b64 load
            v2f a[2], bb[4];
            #pragma unroll
            for (int mi = 0; mi < 2; ++mi)
                a[mi] = *(const v2f*)&As[waveM * 32 + mi * 16 + lm][kk + 2 * lh];
            #pragma unroll
            for (int g = 0; g < 4; ++g)
                bb[g] = *(const v2f*)&Bt[g][waveN * 16 + lm][kk + 2 * lh];
            #pragma unroll
            for (int g = 0; g < 4; ++g)
                #pragma unroll
                for (int mi = 0; mi < 2; ++mi)
                    acc[g][mi] = __builtin_amdgcn_wmma_f32_16x16x4_f32(
                        false, a[mi], false, bb[g], (short)0, acc[g][mi],
                        false, false);
        }
        __syncthreads();
    }

    // --- fused epilogue: all 4 gates in-wave -> f,i,g,o -> c,h --------------
    // C layout: lane holds col n = lane&15, rows (lane>>4)*8 + v, v=0..7
    const int o = o0 + waveN * 16 + lm;
    #pragma unroll
    for (int mi = 0; mi < 2; ++mi) {
        const int rbase = m0 + waveM * 32 + mi * 16 + (lane >> 4) * 8;
        #pragma unroll
        for (int v = 0; v < 8; ++v) {
            int r = rbase + v;
            int t = r >> 6;
            int b = r & (BATCH - 1);
            float gf = acc[0][mi][v] + hterm_l[(b * 4 + 0) * HID + o];
            float gi = acc[1][mi][v] + hterm_l[(b * 4 + 1) * HID + o];
            float gg = acc[2][mi][v] + hterm_l[(b * 4 + 2) * HID + o];
            float go = acc[3][mi][v] + hterm_l[(b * 4 + 3) * HID + o];
            float fg = 1.f / (1.f + expf(-gf));
            float ig = 1.f / (1.f + expf(-gi));
            float g_ = tanhf(gg);
            float og = 1.f / (1.f + expf(-go));
            float c = fg * c0_l[b * HID + o] + ig * g_;
            float h = og * tanhf(c);
            size_t hn = TBH + ((size_t)(t * NLAYER + layer) * BATCH + b) * HID + o;
            out[hn] = h;                    // h_n[t, layer, b, o]
            out[hn + TLBH] = c;             // c_n[t, layer, b, o]
            if (layer == NLAYER - 1)
                out[((size_t)t * BATCH + b) * HID + o] = h;   // x = last-layer h
        }
    }
}

// ---------------------------------------------------------------------------
extern "C" void kernel_launch(void* const* d_in, const int* in_sizes, int n_in,
                              void* d_out, int out_size, void* d_ws, size_t ws_size,
                              hipStream_t stream) {
    const float* inputs = (const float*)d_in[0];   // [T,B,D]
    const float* h0     = (const float*)d_in[1];   // [L,B,H]
    const float* c0     = (const float*)d_in[2];   // [L,B,H]
    const float* W      = (const float*)d_in[3];   // [L,4,D+H,H]
    const float* bias   = (const float*)d_in[4];   // [L,4,H]
    float* out = (float*)d_out;
    float* hterm = (float*)d_ws;                   // [L,B,4,H] = 512 KB

    lstm_hterm<<<(NLAYER * BATCH * 4 * HID) / 256, 256, 0, stream>>>(h0, W, bias, hterm);

    dim3 grid(MTOT / MT, HID / OT);                // (2048, 4)
    // layer 0: A = inputs, rows at t*(B*D) + b*D
    lstm_layer_wmma<<<grid, 256, 0, stream>>>(
        inputs, BATCH * DIMIN, W, hterm, c0, out, 0);
    // layer 1: A = h_n[:,0] inside d_out, rows at t*(L*B*H) + b*H
    lstm_layer_wmma<<<grid, 256, 0, stream>>>(
        out + TBH, NLAYER * BATCH * HID,
        W + (size_t)4 * (DIMIN + HID) * HID,
        hterm + BATCH * 4 * HID, c0 + (size_t)BATCH * HID, out, 1);
}